// OneKeyAttation_56487409877273
// MI455X (gfx1250) — compile-verified
//
#include <hip/hip_runtime.h>
#include <hip/hip_bf16.h>

// ---------------------------------------------------------------------------
// OneKeyAttation for MI455X (gfx1250, wave32).
//
// Key algebraic facts exploited:
//  * softmax(sim,axis=1).mean(axis=1) == 1/12 exactly (softmax sums to 1),
//    so assp_weighted = updated_features * (1/12). The 8.6-GFLOP key conv
//    and sim GEMM are dead code for output 0.
//  * protos_out = q = query_w . prototypes + query_b (tiny GEMM).
//  * nearest-proto selection: argmin ||p-f||^2 = argmin(||p||^2 - 2 p.f),
//    i.e. a (12x256)x(256xNpix) GEMM + per-column argmin.
//
// Pipeline (all fp32, matching reference precision):
//  1. init_map: map[b][pix] = -1.
//  2. assign_kernel: WMMA f32 16x16x4 GEMM of protos x gathered features in
//     LDS (async global->LDS fill, ASYNCcnt-tracked), argmin -> map.
//  3. q_kernel: WMMA f32 16x16x4 for q; writes protos_out.
//  4. out_kernel: bandwidth-bound stream with non-temporal load/store policy
//     (touch-once data must not rinse the 192MB L2; the reused 512KB map and
//     96KB prototype tables stay regular-temporal). ~268 MB => ~11.5us.
// ---------------------------------------------------------------------------

typedef __attribute__((ext_vector_type(2))) float v2f;
typedef __attribute__((ext_vector_type(4))) float v4f;
typedef __attribute__((ext_vector_type(8))) float v8f;
typedef __attribute__((ext_vector_type(4))) int   v4i;

#define BATCH 8
#define CH    256
#define HW    16384   // 128*128
#define KP    12      // prototypes per batch

#if defined(__gfx1250__) && __has_builtin(__builtin_amdgcn_global_load_async_to_lds_b32)
#define HAVE_ASYNC_LDS 1
typedef __attribute__((address_space(1))) int gbl_int_t;   // clang prints as __device__ int*
typedef __attribute__((address_space(3))) int lds_int_t;   // LDS int*
#else
#define HAVE_ASYNC_LDS 0
#endif

// ---------------------------------------------------------------------------
__global__ void init_map_kernel(int* __restrict__ map, int n) {
    int i = blockIdx.x * blockDim.x + threadIdx.x;
    if (i < n) map[i] = -1;
}

// ---------------------------------------------------------------------------
// One block = one batch x 64 gathered pixels. 128 threads = 4 waves.
// S(16x64) = Ppad(16x256) x Fgather(256x64) via v_wmma_f32_16x16x4_f32.
__global__ void assign_kernel(const float* __restrict__ protos,   // (B,12,256)
                              const float* __restrict__ feats,    // (B,256,HW)
                              const int*   __restrict__ pix_idx,  // (n_pix)
                              const int*   __restrict__ flag,     // DomainTrain
                              int n_pix,
                              int* __restrict__ map) {            // (B,HW)
    if (*flag == 0) return;

    extern __shared__ float sm[];
    float* sP   = sm;                 // 16*256 protos (rows 12..15 zero)
    float* sF   = sP + 16 * CH;       // 256*64 gathered feature columns
    float* sS   = sF + CH * 64;       // 16*64 scores
    float* sPn  = sS + 16 * 64;       // 16 proto squared norms
    int*   sPix = (int*)(sPn + 16);   // 64 pixel indices

    const int b    = blockIdx.y;
    const int tile = blockIdx.x;
    const int tid  = threadIdx.x;     // 0..127

    // Stage prototypes (rows 12..15 zero-padded; ignored in argmin).
    for (int idx = tid; idx < 16 * CH; idx += blockDim.x) {
        int k = idx >> 8, c = idx & (CH - 1);
        sP[idx] = (k < KP) ? protos[((size_t)b * KP + k) * CH + c] : 0.0f;
    }
    if (tid < 64) {
        int n = tile * 64 + tid;
        sPix[tid] = pix_idx[(n < n_pix) ? n : (n_pix - 1)];
    }
    __syncthreads();

    // ||p_k||^2
    if (tid < 16) {
        float s = 0.0f;
        for (int c = 0; c < CH; ++c) { float p = sP[tid * CH + c]; s += p * p; }
        sPn[tid] = s;
    }

    // Gather 64 feature columns (256 rows each) into LDS.
#if HAVE_ASYNC_LDS
    for (int idx = tid; idx < CH * 64; idx += blockDim.x) {
        int c = idx >> 6, j = idx & 63;
        const float* gsrc = feats + ((size_t)b * CH + c) * HW + sPix[j];
        __builtin_amdgcn_global_load_async_to_lds_b32(
            (gbl_int_t*)gsrc, (lds_int_t*)(sF + idx), 0, 0);
    }
#if __has_builtin(__builtin_amdgcn_s_wait_asynccnt)
    __builtin_amdgcn_s_wait_asynccnt(0);
#else
    asm volatile("s_wait_asynccnt 0x0" ::: "memory");
#endif
#else
    for (int idx = tid; idx < CH * 64; idx += blockDim.x) {
        int c = idx >> 6, j = idx & 63;
        sF[idx] = feats[((size_t)b * CH + c) * HW + sPix[j]];
    }
#endif
    __syncthreads();

    // WMMA: each wave owns a 16x16 column tile of S.
    const int wave = tid >> 5, lane = tid & 31;
    const int l15 = lane & 15, h = lane >> 4;   // h selects the K pair
    const int n0 = wave * 16;
    v8f acc = {};
    for (int kk = 0; kk < CH; kk += 4) {
        v2f a, bb;
        // A 16x4 f32: VGPR0 = K(2h), VGPR1 = K(2h+1)
        a.x = sP[l15 * CH + kk + 2 * h];
        a.y = sP[l15 * CH + kk + 2 * h + 1];
        // B 4x16 f32 (row striped across lanes)
        bb.x = sF[(kk + 2 * h) * 64 + n0 + l15];
        bb.y = sF[(kk + 2 * h + 1) * 64 + n0 + l15];
        acc = __builtin_amdgcn_wmma_f32_16x16x4_f32(
            false, a, false, bb, (short)0, acc, false, false);
    }
#pragma unroll
    for (int r = 0; r < 8; ++r) {
        int row = r + 8 * h;                    // C/D layout: M=r / M=r+8
        sS[row * 64 + n0 + l15] = acc[r];
    }
    __syncthreads();

    // Per-pixel argmin over 12 prototypes (first-min tie break like argmin).
    if (tid < 64) {
        int n = tile * 64 + tid;
        if (n < n_pix) {
            float best = sPn[0] - 2.0f * sS[tid];
            int bk = 0;
            for (int k = 1; k < KP; ++k) {
                float d = sPn[k] - 2.0f * sS[k * 64 + tid];
                if (d < best) { best = d; bk = k; }
            }
            map[(size_t)b * HW + sPix[tid]] = bk;
        }
    }
}

// ---------------------------------------------------------------------------
// q[b,cls,j,:] = query_w[cls] (128x256) x protos[b,cls,j,:] + query_b[cls].
// One block per (b,cls); 4 waves, each wave 2 M-tiles; N padded 2->16.
__global__ void q_kernel(const float* __restrict__ protos,  // (B,12,256)
                         const float* __restrict__ qw,      // (6,128,256)
                         const float* __restrict__ qb,      // (6,128)
                         float* __restrict__ qout) {        // (B,12,128)
    const int b   = blockIdx.x;
    const int cls = blockIdx.y;
    const int tid = threadIdx.x;                // 0..127
    const int wave = tid >> 5, lane = tid & 31;
    const int l15 = lane & 15, h = lane >> 4;

    const float* W = qw + (size_t)cls * 128 * CH;
    v8f acc[2] = {v8f{}, v8f{}};

    for (int kk = 0; kk < CH; kk += 4) {
        v2f bb; bb.x = 0.0f; bb.y = 0.0f;
        if (l15 < 2) {
            const float* pr = protos + ((size_t)b * KP + cls * 2 + l15) * CH;
            bb.x = pr[kk + 2 * h];
            bb.y = pr[kk + 2 * h + 1];
        }
#pragma unroll
        for (int t = 0; t < 2; ++t) {
            int m = (wave * 2 + t) * 16 + l15;
            v2f a;
            a.x = W[(size_t)m * CH + kk + 2 * h];
            a.y = W[(size_t)m * CH + kk + 2 * h + 1];
            acc[t] = __builtin_amdgcn_wmma_f32_16x16x4_f32(
                false, a, false, bb, (short)0, acc[t], false, false);
        }
    }
#pragma unroll
    for (int t = 0; t < 2; ++t) {
#pragma unroll
        for (int r = 0; r < 8; ++r) {
            int row = (wave * 2 + t) * 16 + r + 8 * h;
            if (l15 < 2) {
                qout[((size_t)b * KP + cls * 2 + l15) * 128 + row] =
                    acc[t][r] + qb[cls * 128 + row];
            }
        }
    }
}

// ---------------------------------------------------------------------------
// out[b,c,pix] = (map[b,pix] >= 0 ? protos[b,map,c] : feats[b,c,pix]) / 12.
// Touch-once traffic is non-temporal; map/protos stay regular (L2-resident).
__global__ void __launch_bounds__(256)
out_kernel(const float* __restrict__ feats,
           const float* __restrict__ protos,
           const int*   __restrict__ map,
           float* __restrict__ out) {
    size_t g = (size_t)blockIdx.x * blockDim.x + threadIdx.x;
    size_t base = g * 4;                        // 4 consecutive pixels
    int pix = (int)(base & (size_t)(HW - 1));
    size_t bc = base >> 14;                     // b*CH + c
    int c = (int)(bc & (CH - 1));
    int b = (int)(bc >> 8);

    const v4f f = __builtin_nontemporal_load((const v4f*)(feats + base));
    const v4i m = *(const v4i*)(map + ((size_t)b << 14) + pix);

    v4f v = f;
    const float* pb = protos + (size_t)b * KP * CH + c;
    if (m.x >= 0) v.x = pb[(size_t)m.x * CH];
    if (m.y >= 0) v.y = pb[(size_t)m.y * CH];
    if (m.z >= 0) v.z = pb[(size_t)m.z * CH];
    if (m.w >= 0) v.w = pb[(size_t)m.w * CH];

    const float s = 1.0f / 12.0f;   // softmax(...).mean over 12 == 1/12
    v *= s;
    __builtin_nontemporal_store(v, (v4f*)(out + base));
}

// ---------------------------------------------------------------------------
extern "C" void kernel_launch(void* const* d_in, const int* in_sizes, int n_in,
                              void* d_out, int out_size, void* d_ws, size_t ws_size,
                              hipStream_t stream) {
    const float* protos  = (const float*)d_in[0];   // (8,6,2,256)
    const float* feats   = (const float*)d_in[1];   // (8,256,128,128)
    const int*   pix_idx = (const int*)d_in[2];     // (n_pix)
    const float* qw      = (const float*)d_in[5];   // (6,128,256)
    const float* qb      = (const float*)d_in[6];   // (6,128)
    const int*   flag    = (const int*)d_in[7];     // DomainTrain
    const int n_pix = in_sizes[2];

    float* out  = (float*)d_out;
    float* qout = out + (size_t)BATCH * CH * HW;    // protos_out after assp_weighted
    int*   map  = (int*)d_ws;                       // (B,HW) int32

    // 1) map = -1
    {
        int n = BATCH * HW;
        init_map_kernel<<<(n + 255) / 256, 256, 0, stream>>>(map, n);
    }
    // 2) nearest-prototype assignment (WMMA GEMM + argmin)
    {
        dim3 grid((n_pix + 63) / 64, BATCH);
        size_t lds = (size_t)(16 * CH + CH * 64 + 16 * 64 + 16) * sizeof(float)
                   + 64 * sizeof(int);
        assign_kernel<<<grid, 128, lds, stream>>>(protos, feats, pix_idx, flag,
                                                  n_pix, map);
    }
    // 3) q / protos_out (WMMA)
    {
        dim3 grid(BATCH, 6);
        q_kernel<<<grid, 128, 0, stream>>>(protos, qw, qb, qout);
    }
    // 4) streaming weighted output
    {
        size_t nvec = (size_t)BATCH * CH * HW / 4;  // 8,388,608
        out_kernel<<<(unsigned)(nvec / 256), 256, 0, stream>>>(feats, protos,
                                                               map, out);
    }
}